// BiLstmCrfNetwork_23012434772470
// MI455X (gfx1250) — compile-verified
//
#include <hip/hip_runtime.h>
#include <math.h>

// CDNA5 (gfx1250) wave32 WMMA f32 16x16x4 operand types
typedef __attribute__((ext_vector_type(2))) float v2f;
typedef __attribute__((ext_vector_type(8))) float v8f;
typedef __attribute__((ext_vector_type(4))) unsigned int v4u;
typedef __attribute__((ext_vector_type(4))) int v4i;
typedef __attribute__((ext_vector_type(8))) int v8i;

#define S_TOK 8192
#define LC    16
#define CHD   100
#define HD    300
#define ED    200
#define FO    20
#define DD    420    // ED + 2*CHD + FO
#define G4C   400    // 4*CHD
#define G4H   1200   // 4*HD
#define NTAG  24

#define U_LDS_DWORDS (G4C * CHD)          // 40000 dwords = 160 KB
#define U_LDS_BYTES  (U_LDS_DWORDS * 4)

__device__ __forceinline__ float sigmoidf_(float x) { return 1.0f / (1.0f + __expf(-x)); }

// ---------------------------------------------------------------------------
// Tensor Data Mover: 1-row tile copy of `ndwords` f32 from global -> LDS.
// D# group0: count=1 | lds_addr | global_addr[56:0] | type=2 ("image")
// D# group1: data_size=4B, tensor_dim0 = tile_dim0 = ndwords, 1 row.
// Wave-level op (EXEC ignored); issue from one wave, wait on TENSORcnt.
// ---------------------------------------------------------------------------
__device__ __forceinline__ void tdm_load_1d_to_lds(const void* gptr,
                                                   unsigned lds_byte_off,
                                                   unsigned ndwords)
{
    const unsigned long long ga = (unsigned long long)(size_t)gptr;
    v4u g0;
    g0.x = 0x1u;                                        // count=1, user descriptor
    g0.y = lds_byte_off;                                // lds_addr (bytes)
    g0.z = (unsigned)(ga & 0xffffffffull);              // global_addr[31:0]
    g0.w = (unsigned)((ga >> 32) & 0x01ffffffull)       // global_addr[56:32]
         | (2u << 30);                                  // type = 2
    v8i g1;
    g1[0] = (int)(2u << 16);                            // data_size = 2 -> 4 bytes
    g1[1] = (int)((ndwords & 0xffffu) << 16);           // tensor_dim0[15:0]
    g1[2] = (int)(((ndwords >> 16) & 0xffffu)           // tensor_dim0[31:16]
         | (1u << 16));                                 // tensor_dim1 = 1
    g1[3] = (int)((ndwords & 0xffffu) << 16);           // tile_dim0
    g1[4] = 1;                                          // tile_dim1 = 1
    g1[5] = (int)ndwords;                               // tensor_dim0_stride[31:0]
    g1[6] = 0;
    g1[7] = 0;
    v4i z4 = {0, 0, 0, 0};
#if __clang_major__ >= 23
    v8i z8 = {0, 0, 0, 0, 0, 0, 0, 0};
    __builtin_amdgcn_tensor_load_to_lds(g0, g1, z4, z4, z8, 0);
#else
    __builtin_amdgcn_tensor_load_to_lds(g0, g1, z4, z4, 0);
#endif
}

// ---------------------------------------------------------------------------
// Char-level BiLSTM, fused: embedding gather + (x@W^T + h@U^T + b) via WMMA
// + gate nonlinearities. One block = 16 tokens, blockIdx.y = direction.
// U (160 KB) is staged once into dynamic LDS by the TDM; A-operands and the
// recurrence B-operand then come from LDS, W streams from L2.
// WMMA f32 16x16x4 layouts (32-bit operands, wave32):
//   A 16x4 : lane&15 = M, VGPR0/1 hold K = 2*(lane>>4) .. +1
//   B 4x16 : lane&15 = N, VGPR0/1 hold K = 2*(lane>>4) .. +1
//   C 16x16: VGPR r holds M = r + 8*(lane>>4), N = lane&15
// ---------------------------------------------------------------------------
__global__ __launch_bounds__(256) void char_lstm_kernel(
    const int* __restrict__ char_ids, const int* __restrict__ char_lengths,
    const float* __restrict__ char_emb,
    const float* __restrict__ Wf, const float* __restrict__ Uf, const float* __restrict__ bf,
    const float* __restrict__ Wb, const float* __restrict__ Ub, const float* __restrict__ bb,
    float* __restrict__ char_vec)
{
    const int dir     = blockIdx.y;
    const int tokBase = blockIdx.x * 16;
    const float* W = dir ? Wb : Wf;
    const float* U = dir ? Ub : Uf;
    const float* B = dir ? bb : bf;

    __shared__ float xt[16][104];
    __shared__ float hh[16][104];
    __shared__ float cc[16][104];
    __shared__ float gg[16][G4C];
    __shared__ int   lens[16];
    extern __shared__ float Ulds[];          // 160 KB dynamic: U[400][100]

    const int tid   = threadIdx.x;
    const int wave  = tid >> 5;
    const int lane  = tid & 31;
    const int lm    = lane & 15;
    const int lhalf = lane >> 4;

    // TDM: stage U -> LDS (dynamic region begins at groupstaticsize)
    if (wave == 0) {
        tdm_load_1d_to_lds(U, __builtin_amdgcn_groupstaticsize(), U_LDS_DWORDS);
    }

    for (int i = tid; i < 16 * 104; i += 256) {
        (&hh[0][0])[i] = 0.0f;
        (&cc[0][0])[i] = 0.0f;
    }
    if (tid < 16) lens[tid] = char_lengths[tokBase + tid];

    __builtin_amdgcn_s_wait_tensorcnt(0);    // issuing wave waits; others pass
    __syncthreads();

    for (int t = 0; t < LC; ++t) {
        const int pos = dir ? (LC - 1 - t) : t;

        // gather x_t = char_emb[char_ids[token][pos]]  (char_emb is L2-resident)
        for (int i = tid; i < 16 * CHD; i += 256) {
            const int m = i / CHD, j = i % CHD;
            const int ci = char_ids[(tokBase + m) * LC + pos];
            xt[m][j] = char_emb[ci * CHD + j];
        }
        __syncthreads();

        // gates[16][400] = x_t @ W^T + h @ U^T + b   via WMMA f32 16x16x4
        for (int nt = wave; nt < G4C / 16; nt += 8) {   // wave-uniform
            const int gn = nt * 16 + lm;                // gate column for this lane
            v8f c;
            const float bv = B[gn];
            #pragma unroll
            for (int r = 0; r < 8; ++r) c[r] = bv;
            const int koff = 2 * lhalf;
            #pragma unroll 5
            for (int kt = 0; kt < CHD / 4; ++kt) {      // x @ W^T (B from L2)
                const int kb = kt * 4 + koff;
                v2f a = *(const v2f*)&xt[lm][kb];
                v2f b = *(const v2f*)&W[gn * CHD + kb];
                c = __builtin_amdgcn_wmma_f32_16x16x4_f32(false, a, false, b, (short)0, c, false, false);
            }
            #pragma unroll 5
            for (int kt = 0; kt < CHD / 4; ++kt) {      // h @ U^T (B from LDS)
                const int kb = kt * 4 + koff;
                v2f a = *(const v2f*)&hh[lm][kb];
                v2f b = *(const v2f*)&Ulds[gn * CHD + kb];
                c = __builtin_amdgcn_wmma_f32_16x16x4_f32(false, a, false, b, (short)0, c, false, false);
            }
            #pragma unroll
            for (int r = 0; r < 8; ++r) {
                const int m = r + 8 * lhalf;
                gg[m][gn] = c[r];
            }
        }
        __syncthreads();

        // LSTM cell update (PyTorch gate order i,f,g,o) with ragged masking
        for (int i = tid; i < 16 * CHD; i += 256) {
            const int m = i / CHD, j = i % CHD;
            if (pos < lens[m]) {
                const float gi = gg[m][j];
                const float gf = gg[m][CHD + j];
                const float gc = gg[m][2 * CHD + j];
                const float go = gg[m][3 * CHD + j];
                const float c2 = sigmoidf_(gf) * cc[m][j] + sigmoidf_(gi) * tanhf(gc);
                const float h2 = sigmoidf_(go) * tanhf(c2);
                cc[m][j] = c2;
                hh[m][j] = h2;
            }
        }
        __syncthreads();
    }

    // final hidden -> char_vec[:, dir*CHD : dir*CHD+CHD]
    for (int i = tid; i < 16 * CHD; i += 256) {
        const int m = i / CHD, j = i % CHD;
        char_vec[(size_t)(tokBase + m) * (2 * CHD) + dir * CHD + j] = hh[m][j];
    }
}

// ---------------------------------------------------------------------------
// tok[s] = [word_emb[token_ids[s]] | char_vec[s] | other_feats[s]]
// ---------------------------------------------------------------------------
__global__ void concat_kernel(const int* __restrict__ token_ids,
                              const float* __restrict__ word_emb,
                              const float* __restrict__ char_vec,
                              const float* __restrict__ other_feats,
                              float* __restrict__ tok)
{
    const int total = S_TOK * DD;
    for (int idx = blockIdx.x * blockDim.x + threadIdx.x; idx < total;
         idx += gridDim.x * blockDim.x) {
        const int s = idx / DD, c = idx % DD;
        float v;
        if (c < ED)               v = word_emb[(size_t)token_ids[s] * ED + c];
        else if (c < ED + 2*CHD)  v = char_vec[(size_t)s * (2 * CHD) + (c - ED)];
        else                      v = other_feats[(size_t)s * FO + (c - ED - 2 * CHD)];
        tok[idx] = v;
    }
}

// ---------------------------------------------------------------------------
// Word-LSTM input precompute: xW[dir] = tok @ wW[dir]^T + b   (WMMA GEMM)
// grid = (S/16, ceil(75/8), 2); one wave per 16-wide N tile, 105 K-iters.
// ---------------------------------------------------------------------------
__global__ __launch_bounds__(256) void word_xw_kernel(
    const float* __restrict__ tok,
    const float* __restrict__ Wf, const float* __restrict__ bf,
    const float* __restrict__ Wb, const float* __restrict__ bb,
    float* __restrict__ xwf, float* __restrict__ xwb)
{
    const int dir = blockIdx.z;
    const float* W = dir ? Wb : Wf;
    const float* B = dir ? bb : bf;
    float* out = dir ? xwb : xwf;

    const int tid = threadIdx.x, wave = tid >> 5, lane = tid & 31;
    const int lm = lane & 15, lhalf = lane >> 4;
    const int mbase = blockIdx.x * 16;
    const int nt = blockIdx.y * 8 + wave;
    if (nt >= G4H / 16) return;        // wave-uniform tail
    const int gn = nt * 16 + lm;

    v8f c;
    const float bv = B[gn];
    #pragma unroll
    for (int r = 0; r < 8; ++r) c[r] = bv;

    const int m = mbase + lm;
    const int koff = 2 * lhalf;
    #pragma unroll 5
    for (int kt = 0; kt < DD / 4; ++kt) {   // K = 420
        const int kb = kt * 4 + koff;
        v2f a = *(const v2f*)&tok[(size_t)m * DD + kb];
        v2f b = *(const v2f*)&W[(size_t)gn * DD + kb];
        c = __builtin_amdgcn_wmma_f32_16x16x4_f32(false, a, false, b, (short)0, c, false, false);
    }
    #pragma unroll
    for (int r = 0; r < 8; ++r) {
        const int mm = mbase + r + 8 * lhalf;
        out[(size_t)mm * G4H + gn] = c[r];
    }
}

// ---------------------------------------------------------------------------
// Word-LSTM recurrence: serial over S steps; one persistent block/direction.
// h,c,g live in LDS; U rows stream from L2 (1.44 MB, fully L2-resident).
// Prefetch next step's xW row so the L2 fetch overlaps this step's matvec.
// Backward direction writes its hidden at the original index -> hbr = hb[::-1].
// ---------------------------------------------------------------------------
__global__ __launch_bounds__(1024) void word_rec_kernel(
    const float* __restrict__ xwf, const float* __restrict__ xwb,
    const float* __restrict__ Uf, const float* __restrict__ Ub,
    float* __restrict__ hf, float* __restrict__ hbr)
{
    const int dir = blockIdx.x;
    const float* xw = dir ? xwb : xwf;
    const float* U  = dir ? Ub  : Uf;
    float* hout = dir ? hbr : hf;

    __shared__ float h[HD];
    __shared__ float cst[HD];
    __shared__ float g[G4H];

    const int tid = threadIdx.x;
    for (int i = tid; i < HD; i += 1024) { h[i] = 0.0f; cst[i] = 0.0f; }
    __syncthreads();

    for (int t = 0; t < S_TOK; ++t) {
        const int s = dir ? (S_TOK - 1 - t) : t;
        const float* xr = xw + (size_t)s * G4H;

        if (t + 1 < S_TOK) {           // overlap next row's L2 fetch
            const int sn = dir ? (S_TOK - 2 - t) : (t + 1);
            __builtin_prefetch(&xw[(size_t)sn * G4H + tid], 0, 1);
        }

        for (int j = tid; j < G4H; j += 1024) {
            float acc = xr[j];
            const float* Ur = U + (size_t)j * HD;
            #pragma unroll 4
            for (int k = 0; k < HD; ++k) acc = fmaf(Ur[k], h[k], acc);
            g[j] = acc;
        }
        __syncthreads();

        for (int j = tid; j < HD; j += 1024) {
            const float c2 = sigmoidf_(g[HD + j]) * cst[j] + sigmoidf_(g[j]) * tanhf(g[2 * HD + j]);
            const float h2 = sigmoidf_(g[3 * HD + j]) * tanhf(c2);
            cst[j] = c2;
            h[j]   = h2;
            hout[(size_t)s * HD + j] = h2;
        }
        __syncthreads();
    }
}

// ---------------------------------------------------------------------------
// Tag projection: out = [hf | hbr] @ tagW^T + tagb  (WMMA, N padded 24->32,
// invalid lanes zeroed via multiplies so EXEC stays all-1s around WMMA)
// ---------------------------------------------------------------------------
__global__ __launch_bounds__(64) void tag_kernel(
    const float* __restrict__ hf, const float* __restrict__ hbr,
    const float* __restrict__ tagW, const float* __restrict__ tagb,
    float* __restrict__ out)
{
    const int tid = threadIdx.x, wave = tid >> 5, lane = tid & 31;
    const int lm = lane & 15, lhalf = lane >> 4;
    const int mbase = blockIdx.x * 16;
    const int gn  = wave * 16 + lm;           // 0..31, valid < 24
    const int gnc = gn < NTAG ? gn : (NTAG - 1);
    const float sel = (gn < NTAG) ? 1.0f : 0.0f;

    v8f c;
    const float bv = tagb[gnc] * sel;
    #pragma unroll
    for (int r = 0; r < 8; ++r) c[r] = bv;

    const int m = mbase + lm;
    const int koff = 2 * lhalf;
    #pragma unroll 5
    for (int kt = 0; kt < (2 * HD) / 4; ++kt) {   // K = 600
        const int kb = kt * 4 + koff;
        const float* arow = (kb < HD) ? &hf[(size_t)m * HD + kb]
                                      : &hbr[(size_t)m * HD + (kb - HD)];
        v2f a = *(const v2f*)arow;
        v2f b = *(const v2f*)&tagW[(size_t)gnc * (2 * HD) + kb];
        b.x *= sel; b.y *= sel;
        c = __builtin_amdgcn_wmma_f32_16x16x4_f32(false, a, false, b, (short)0, c, false, false);
    }
    if (gn < NTAG) {
        #pragma unroll
        for (int r = 0; r < 8; ++r) {
            const int mm = mbase + r + 8 * lhalf;
            out[(size_t)mm * NTAG + gn] = c[r];
        }
    }
}

// ---------------------------------------------------------------------------
extern "C" void kernel_launch(void* const* d_in, const int* in_sizes, int n_in,
                              void* d_out, int out_size, void* d_ws, size_t ws_size,
                              hipStream_t stream)
{
    (void)in_sizes; (void)n_in; (void)out_size; (void)ws_size;

    const int*   token_ids   = (const int*)  d_in[0];
    const int*   char_ids    = (const int*)  d_in[1];
    const int*   char_len    = (const int*)  d_in[2];
    const float* other_feats = (const float*)d_in[3];
    const float* word_emb    = (const float*)d_in[4];
    const float* char_emb    = (const float*)d_in[5];
    const float* cWf = (const float*)d_in[6],  *cUf = (const float*)d_in[7],  *cbf = (const float*)d_in[8];
    const float* cWb = (const float*)d_in[9],  *cUb = (const float*)d_in[10], *cbb = (const float*)d_in[11];
    const float* wWf = (const float*)d_in[12], *wUf = (const float*)d_in[13], *wbf = (const float*)d_in[14];
    const float* wWb = (const float*)d_in[15], *wUb = (const float*)d_in[16], *wbb = (const float*)d_in[17];
    const float* tagW = (const float*)d_in[18], *tagb = (const float*)d_in[19];
    float* out = (float*)d_out;

    // workspace layout (f32), total ~119 MB
    char* ws = (char*)d_ws;
    size_t o = 0;
    float* char_vec = (float*)(ws + o); o += (size_t)S_TOK * 2 * CHD * 4;
    float* tok      = (float*)(ws + o); o += (size_t)S_TOK * DD * 4;
    float* xwf      = (float*)(ws + o); o += (size_t)S_TOK * G4H * 4;
    float* xwb      = (float*)(ws + o); o += (size_t)S_TOK * G4H * 4;
    float* hfb      = (float*)(ws + o); o += (size_t)S_TOK * HD * 4;
    float* hbr      = (float*)(ws + o); o += (size_t)S_TOK * HD * 4;

    char_lstm_kernel<<<dim3(S_TOK / 16, 2), 256, U_LDS_BYTES, stream>>>(
        char_ids, char_len, char_emb, cWf, cUf, cbf, cWb, cUb, cbb, char_vec);

    concat_kernel<<<dim3(1024), 256, 0, stream>>>(
        token_ids, word_emb, char_vec, other_feats, tok);

    word_xw_kernel<<<dim3(S_TOK / 16, 10, 2), 256, 0, stream>>>(
        tok, wWf, wbf, wWb, wbb, xwf, xwb);

    word_rec_kernel<<<dim3(2), 1024, 0, stream>>>(
        xwf, xwb, wUf, wUb, hfb, hbr);

    tag_kernel<<<dim3(S_TOK / 16), 64, 0, stream>>>(
        hfb, hbr, tagW, tagb, out);
}